// LightGCN_76055280877683
// MI455X (gfx1250) — compile-verified
//
#include <hip/hip_runtime.h>
#include <math.h>

typedef __attribute__((ext_vector_type(2))) float v2f;
typedef __attribute__((ext_vector_type(8))) float v8f;

#define DIM 64
#define NLAYER 3

// ---------------- utility ----------------
__global__ void k_zero(float* p, long n) {
    long i = (long)blockIdx.x * blockDim.x + threadIdx.x;
    if (i < n) p[i] = 0.0f;
}

// ---------------- degree / norm ----------------
__global__ void k_degree(const int* __restrict__ dst, float* __restrict__ deg, int E) {
    int e = blockIdx.x * blockDim.x + threadIdx.x;
    if (e < E) atomicAdd(&deg[dst[e]], 1.0f);
}

__global__ void k_dinv(const float* __restrict__ deg, float* __restrict__ dinv, int N) {
    int i = blockIdx.x * blockDim.x + threadIdx.x;
    if (i < N) {
        float d = deg[i];
        dinv[i] = (d > 0.0f) ? rsqrtf(d) : 0.0f;
    }
}

// ---------------- soft-thresholded sparse embedding ----------------
__global__ void k_sparse(const float* __restrict__ weight, const float* __restrict__ s,
                         float* __restrict__ h0, float* __restrict__ acc, long n) {
    long i = (long)blockIdx.x * blockDim.x + threadIdx.x;
    if (i >= n) return;
    float thresh = 1.0f / (1.0f + __expf(-s[0]));
    float w = weight[i];
    float v = fmaxf(fabsf(w) - thresh, 0.0f);
    v = copysignf(v, w);            // v==0 when |w|<=thresh, so sign(0) case is safe
    h0[i] = v;
    acc[i] = v;
}

// ---------------- SpMM: one edge per wave32, float2 per lane ----------------
__global__ __launch_bounds__(256) void k_spmm(const int* __restrict__ src,
                                              const int* __restrict__ dst,
                                              const float* __restrict__ dinv,
                                              const float* __restrict__ hin,
                                              float* __restrict__ hout, int E) {
    int gid = blockIdx.x * blockDim.x + threadIdx.x;
    int e = gid >> 5;
    int lane = threadIdx.x & 31;
    if (e >= E) return;
    int si = src[e], di = dst[e];
    float a = dinv[si] * dinv[di];
    const float2 v = *(const float2*)(hin + (size_t)si * DIM + lane * 2);
    float* o = hout + (size_t)di * DIM + lane * 2;
    atomicAdd(o, v.x * a);
    atomicAdd(o + 1, v.y * a);
}

__global__ void k_accadd(float* __restrict__ acc, const float* __restrict__ h, long n) {
    long i = (long)blockIdx.x * blockDim.x + threadIdx.x;
    if (i < n) acc[i] += h[i];
}

// ---------------- BPR scoring via V_WMMA_F32_16X16X4_F32 ----------------
// Wave w handles 16 batch rows. Computes full 16x16 tiles U*P^T and U*N^T
// via a K-swept chain of 16x16x4 f32 WMMA ops, extracts the diagonal
// (rowwise dot products) through an LDS bounce, accumulates softplus loss.
__global__ __launch_bounds__(256) void k_bpr(const float* __restrict__ acc,
                                             const int* __restrict__ users,
                                             const int* __restrict__ pos,
                                             const int* __restrict__ neg,
                                             float* __restrict__ accum) {
    __shared__ float ldsP[8 * 256];
    __shared__ float ldsN[8 * 256];
    int wl   = threadIdx.x >> 5;       // wave in block (8 waves)
    int lane = threadIdx.x & 31;
    int gw   = blockIdx.x * 8 + wl;    // global wave -> 16 rows
    int r    = lane & 15;              // row within tile for this lane
    int half = lane >> 4;              // lanes 16-31 hold K+2,K+3
    int row  = gw * 16 + r;

    const float* au = acc + (size_t)users[row] * DIM;
    const float* ap = acc + (size_t)pos[row]   * DIM;
    const float* an = acc + (size_t)neg[row]   * DIM;

    v8f cp = {0.f, 0.f, 0.f, 0.f, 0.f, 0.f, 0.f, 0.f};
    v8f cn = {0.f, 0.f, 0.f, 0.f, 0.f, 0.f, 0.f, 0.f};

#pragma unroll
    for (int kk = 0; kk < 16; ++kk) {
        int k = kk * 4 + half * 2;
        v2f a  = { au[k], au[k + 1] };   // A = U[:, k:k+4]    (16x4, row per lane)
        v2f bp = { ap[k], ap[k + 1] };   // B = P[:, k:k+4]^T  (4x16)
        v2f bn = { an[k], an[k + 1] };
        cp = __builtin_amdgcn_wmma_f32_16x16x4_f32(false, a, false, bp, (short)0, cp, false, false);
        cn = __builtin_amdgcn_wmma_f32_16x16x4_f32(false, a, false, bn, (short)0, cn, false, false);
    }

#pragma unroll
    for (int q = 0; q < 8; ++q) {
        ldsP[wl * 256 + lane * 8 + q] = cp[q];
        ldsN[wl * 256 + lane * 8 + q] = cn[q];
    }
    __syncthreads();

    if (lane < 16) {
        // diagonal (m,m): m<8 -> VGPR m on lane m; m>=8 -> VGPR m-8 on lane m+16
        int sl = (lane < 8) ? lane : lane + 16;
        int q  = lane & 7;
        float ps = ldsP[wl * 256 + sl * 8 + q] * 0.0625f;  // (acc/4).(acc/4) = dot/16
        float ns = ldsN[wl * 256 + sl * 8 + q] * 0.0625f;
        float x  = ns - ps;
        float sp = (x > 0.0f) ? x + log1pf(__expf(-x)) : log1pf(__expf(x));
        atomicAdd(&accum[0], sp);
    }
}

// ---------------- regularizer: sum of squares of ego embeddings ----------------
__global__ __launch_bounds__(256) void k_reg(const float* __restrict__ weight,
                                             const int* __restrict__ users,
                                             const int* __restrict__ pos,
                                             const int* __restrict__ neg,
                                             float* __restrict__ accum, int B) {
    int t = blockIdx.x * blockDim.x + threadIdx.x;
    if (t >= B * DIM) return;
    int row = t >> 6, c = t & 63;
    float a = weight[(size_t)users[row] * DIM + c];
    float b = weight[(size_t)pos[row]   * DIM + c];
    float d = weight[(size_t)neg[row]   * DIM + c];
    float v = a * a + b * b + d * d;
#pragma unroll
    for (int off = 16; off > 0; off >>= 1) v += __shfl_down(v, off, 32);
    if ((threadIdx.x & 31) == 0) atomicAdd(&accum[1], v);
}

// ---------------- finalize ----------------
__global__ void k_final(const float* __restrict__ accum, const float* __restrict__ s,
                        float* __restrict__ out, int B) {
    if (threadIdx.x != 0 || blockIdx.x != 0) return;
    float b = (float)B;
    float loss_emb = accum[0] / b;
    float reg_loss = 0.5f * accum[1] / b * 1e-4f;
    float s_loss   = 0.5f * s[0] * s[0] / b * 1e-4f;
    out[0] = loss_emb + reg_loss + s_loss;
    out[1] = loss_emb;
    out[2] = reg_loss;
    out[3] = s_loss;
}

static inline int cdiv(long a, long b) { return (int)((a + b - 1) / b); }

extern "C" void kernel_launch(void* const* d_in, const int* in_sizes, int n_in,
                              void* d_out, int out_size, void* d_ws, size_t ws_size,
                              hipStream_t stream) {
    const float* weight = (const float*)d_in[0];
    const float* s      = (const float*)d_in[1];
    const int*   src    = (const int*)d_in[2];
    const int*   dst    = (const int*)d_in[3];
    const int*   users  = (const int*)d_in[4];
    const int*   pos    = (const int*)d_in[5];
    const int*   neg    = (const int*)d_in[6];

    int N = in_sizes[0] / DIM;     // 100000
    int E = in_sizes[2];           // 1600000
    int B = in_sizes[4];           // 4096
    long nd = (long)N * DIM;

    // workspace layout (floats): accum[16] | deg[N] | dinv[N] | h0[nd] | h1[nd] | acc[nd]
    float* ws    = (float*)d_ws;
    float* accum = ws;
    float* deg   = ws + 16;
    float* dinv  = deg + N;
    float* h0    = dinv + N;
    float* h1    = h0 + nd;
    float* acc   = h1 + nd;

    // zero accumulators + degree (harness poisons d_ws)
    k_zero<<<cdiv(16 + N, 256), 256, 0, stream>>>(accum, 16 + (long)N);
    k_degree<<<cdiv(E, 256), 256, 0, stream>>>(dst, deg, E);
    k_dinv<<<cdiv(N, 256), 256, 0, stream>>>(deg, dinv, N);
    k_sparse<<<cdiv(nd, 256), 256, 0, stream>>>(weight, s, h0, acc, nd);

    float* cur = h0;
    float* nxt = h1;
    for (int l = 0; l < NLAYER; ++l) {
        k_zero<<<cdiv(nd, 256), 256, 0, stream>>>(nxt, nd);
        k_spmm<<<cdiv((long)E * 32, 256), 256, 0, stream>>>(src, dst, dinv, cur, nxt, E);
        k_accadd<<<cdiv(nd, 256), 256, 0, stream>>>(acc, nxt, nd);
        float* t = cur; cur = nxt; nxt = t;
    }

    // 4096 rows -> 256 waves -> 32 blocks of 8 waves
    k_bpr<<<B / 128, 256, 0, stream>>>(acc, users, pos, neg, accum);
    k_reg<<<cdiv((long)B * DIM, 256), 256, 0, stream>>>(weight, users, pos, neg, accum, B);
    k_final<<<1, 1, 0, stream>>>(accum, s, (float*)d_out, B);
}